// BondGCNLayer_19997367730282
// MI455X (gfx1250) — compile-verified
//
#include <hip/hip_runtime.h>

// BondGCN layer for MI455X (gfx1250, wave32).
// Pass 1 (stats): per 16-edge tile, e = edge_attr@W0^T + x[src]@W1^T + x[dst]@W2^T + biases
//   via 12x V_WMMA_F32_16X16X4_F32; accumulate per-feature sum / sum^2 with a
//   deterministic shuffle+LDS tree into per-block partials.
// Pass 2 (reduce): fold 1024 block partials -> A[f] = gamma*rsqrt(var+eps), B[f] = beta - mean*A.
// Pass 3 (apply): recompute e (cheaper than spilling 205MB), out = edge_attr + relu(e*A + B).
// x (6.4MB) + weights are L2-resident; edge_attr/out streamed with non-temporal hints.

typedef __attribute__((ext_vector_type(2))) float v2f;
typedef __attribute__((ext_vector_type(8))) float v8f;

#define TPB 256
#define WPB (TPB / 32)       // waves per block
#define NBLK_STATS 1024      // partials: 1024*32 floats = 128KB in d_ws

struct Bfrag { v2f k[4]; };

__device__ __forceinline__ v2f ldv2(const float* p) { return *(const v2f*)p; }
__device__ __forceinline__ v2f ldv2_nt(const float* p) {
    return __builtin_nontemporal_load((const v2f*)p);
}

// B fragment of W^T (B[k][n] = W[n][k]); lane L<16: rows n=L, K-pair k0+{0,1};
// lane L>=16: rows n=L-16, K-pair k0+{2,3}. Same addressing as the A fragment.
__device__ __forceinline__ Bfrag load_bfrag(const float* __restrict__ W, int row, int colb) {
    Bfrag f;
#pragma unroll
    for (int k = 0; k < 4; ++k) f.k[k] = ldv2(W + row * 16 + colb + 4 * k);
    return f;
}

// Compute one 16-edge tile's e fragment (pre-BN): 12 WMMA f32 16x16x4 + bias.
__device__ __forceinline__ v8f compute_e(
    int base, int row, int colb, int NE,
    const float* __restrict__ ea, const int* __restrict__ ei,
    const float* __restrict__ x,
    const Bfrag& B0, const Bfrag& B1, const Bfrag& B2, float biasSum)
{
    const int e = base + row;
    const int s = ei[e];            // src node
    const int d = ei[NE + e];       // dst node
    const float* pe = ea + (size_t)e * 16 + colb;   // streaming (HBM)
    const float* ps = x  + (size_t)s * 16 + colb;   // gather (L2-resident)
    const float* pd = x  + (size_t)d * 16 + colb;
    v8f c = {0.f, 0.f, 0.f, 0.f, 0.f, 0.f, 0.f, 0.f};
#pragma unroll
    for (int k = 0; k < 4; ++k) {
        v2f aE = ldv2_nt(pe + 4 * k);
        v2f aS = ldv2(ps + 4 * k);
        v2f aD = ldv2(pd + 4 * k);
        c = __builtin_amdgcn_wmma_f32_16x16x4_f32(false, aE, false, B0.k[k], (short)0, c, false, false);
        c = __builtin_amdgcn_wmma_f32_16x16x4_f32(false, aS, false, B1.k[k], (short)0, c, false, false);
        c = __builtin_amdgcn_wmma_f32_16x16x4_f32(false, aD, false, B2.k[k], (short)0, c, false, false);
    }
#pragma unroll
    for (int v = 0; v < 8; ++v) c[v] += biasSum;   // feature = row for every c[v]
    return c;
}

__global__ void __launch_bounds__(TPB)
stats_kernel(const float* __restrict__ ea, const int* __restrict__ ei,
             const float* __restrict__ x,
             const float* __restrict__ W0, const float* __restrict__ b0,
             const float* __restrict__ W1, const float* __restrict__ b1,
             const float* __restrict__ W2, const float* __restrict__ b2,
             float* __restrict__ partials, int NE, int ntiles)
{
    __shared__ float lds[WPB * 32];
    const int lane  = threadIdx.x & 31;
    const int wib   = threadIdx.x >> 5;
    const int gwave = (blockIdx.x * TPB + threadIdx.x) >> 5;
    const int nwav  = (gridDim.x * TPB) >> 5;
    const int row   = lane & 15;
    const int colb  = (lane >> 4) * 2;

    const Bfrag B0 = load_bfrag(W0, row, colb);
    const Bfrag B1 = load_bfrag(W1, row, colb);
    const Bfrag B2 = load_bfrag(W2, row, colb);
    const float biasSum = b0[row] + b1[row] + b2[row];

    float accS = 0.f, accQ = 0.f;
    for (int t = gwave; t < ntiles; t += nwav) {           // uniform per wave: EXEC stays all-1
        v8f c = compute_e(t * 16, row, colb, NE, ea, ei, x, B0, B1, B2, biasSum);
#pragma unroll
        for (int v = 0; v < 8; ++v) { accS += c[v]; accQ += c[v] * c[v]; }
    }
    // Lane L and L+16 hold the same feature (row) over disjoint edges: fold them.
    accS += __shfl_xor(accS, 16, 32);
    accQ += __shfl_xor(accQ, 16, 32);
    if (lane < 16) {
        lds[wib * 32 + lane]      = accS;
        lds[wib * 32 + 16 + lane] = accQ;
    }
    __syncthreads();
    if (threadIdx.x < 32) {                                // fixed-order (deterministic) fold
        float v = 0.f;
#pragma unroll
        for (int w = 0; w < WPB; ++w) v += lds[w * 32 + threadIdx.x];
        partials[blockIdx.x * 32 + threadIdx.x] = v;       // [0..15]=sum, [16..31]=sumsq
    }
}

__global__ void reduce_kernel(const float* __restrict__ partials,
                              float* __restrict__ stats,
                              const float* __restrict__ gamma,
                              const float* __restrict__ beta,
                              int nblk, float invNE)
{
    const int tid = threadIdx.x;   // 32 threads, one wave
    float acc = 0.f;
    for (int b = 0; b < nblk; ++b) acc += partials[b * 32 + tid];
    const float other = __shfl_xor(acc, 16, 32);  // lanes 0..15: their sumsq
    if (tid < 16) {
        const float mean = acc * invNE;
        const float var  = other * invNE - mean * mean;   // biased var, matches torch BN
        const float inv  = rsqrtf(var + 1e-5f);
        const float A    = gamma[tid] * inv;
        stats[tid]      = A;
        stats[16 + tid] = beta[tid] - mean * A;
    }
}

__global__ void __launch_bounds__(TPB)
apply_kernel(const float* __restrict__ ea, const int* __restrict__ ei,
             const float* __restrict__ x,
             const float* __restrict__ W0, const float* __restrict__ b0,
             const float* __restrict__ W1, const float* __restrict__ b1,
             const float* __restrict__ W2, const float* __restrict__ b2,
             const float* __restrict__ stats, float* __restrict__ out,
             int NE, int ntiles)
{
    const int lane  = threadIdx.x & 31;
    const int gwave = (blockIdx.x * TPB + threadIdx.x) >> 5;
    if (gwave >= ntiles) return;                 // wave-uniform branch
    const int row  = lane & 15;
    const int hi   = lane >> 4;
    const int colb = hi * 2;

    const Bfrag B0 = load_bfrag(W0, row, colb);
    const Bfrag B1 = load_bfrag(W1, row, colb);
    const Bfrag B2 = load_bfrag(W2, row, colb);
    const float biasSum = b0[row] + b1[row] + b2[row];
    const float A  = stats[row];
    const float Bb = stats[16 + row];

    const int base = gwave * 16;
    v8f c = compute_e(base, row, colb, NE, ea, ei, x, B0, B1, B2, biasSum);

#pragma unroll
    for (int v = 0; v < 8; ++v) {
        const int m = base + v + 8 * hi;               // edge index for this c slot
        const size_t off = (size_t)m * 16 + row;       // feature = row
        const float e = c[v] * A + Bb;
        const float r = e > 0.f ? e : 0.f;
        const float resid = __builtin_nontemporal_load(ea + off);
        __builtin_nontemporal_store(resid + r, out + off);
    }
}

extern "C" void kernel_launch(void* const* d_in, const int* in_sizes, int n_in,
                              void* d_out, int out_size, void* d_ws, size_t ws_size,
                              hipStream_t stream)
{
    const float* x     = (const float*)d_in[0];
    const int*   ei    = (const int*)d_in[1];     // edge_index (2, NE), int32
    const float* ea    = (const float*)d_in[2];   // edge_attr (NE, 16)
    const float* W0    = (const float*)d_in[3];
    const float* b0    = (const float*)d_in[4];
    const float* W1    = (const float*)d_in[5];
    const float* b1    = (const float*)d_in[6];
    const float* W2    = (const float*)d_in[7];
    const float* b2    = (const float*)d_in[8];
    const float* gamma = (const float*)d_in[9];
    const float* beta  = (const float*)d_in[10];
    float* out = (float*)d_out;

    const int NE     = in_sizes[2] / 16;   // 3,200,000
    const int ntiles = NE / 16;            // 200,000 (exact)

    float* partials = (float*)d_ws;                  // NBLK_STATS*32 floats
    float* stats    = partials + NBLK_STATS * 32;    // 32 floats: A[16], B[16]

    stats_kernel<<<NBLK_STATS, TPB, 0, stream>>>(ea, ei, x, W0, b0, W1, b1, W2, b2,
                                                 partials, NE, ntiles);
    reduce_kernel<<<1, 32, 0, stream>>>(partials, stats, gamma, beta,
                                        NBLK_STATS, 1.0f / (float)NE);
    const int nblocks = (ntiles + WPB - 1) / WPB;
    apply_kernel<<<nblocks, TPB, 0, stream>>>(ea, ei, x, W0, b0, W1, b1, W2, b2,
                                              stats, out, NE, ntiles);
}